// SAGENet_ray_1769526526168
// MI455X (gfx1250) — compile-verified
//
#include <hip/hip_runtime.h>
#include <math.h>

// ---------------------------------------------------------------------------
// SAGENet on MI455X (gfx1250, wave32).
// Pipeline per layer: scatter-mean over edges -> dual GEMM via
// V_WMMA_F32_16X16X4_F32 -> BatchNorm(+ReLU). Final layer is a 128->1 dot
// product fused with Linear(1,1) + sigmoid.
// ---------------------------------------------------------------------------

typedef float v2f __attribute__((ext_vector_type(2)));
typedef float v8f __attribute__((ext_vector_type(8)));

#define N_NODES 100000

// -------------------- degree / inverse-degree ------------------------------
__global__ void degree_kernel(const int* __restrict__ dst, float* __restrict__ cnt, int E) {
    int i = blockIdx.x * blockDim.x + threadIdx.x;
    int stride = gridDim.x * blockDim.x;
    for (; i < E; i += stride) atomicAdd(&cnt[dst[i]], 1.0f);
}

__global__ void invdeg_kernel(float* __restrict__ cnt, int N) {
    int i = blockIdx.x * blockDim.x + threadIdx.x;
    if (i < N) cnt[i] = 1.0f / fmaxf(cnt[i], 1.0f);
}

// -------------------- edge scatter-add (wave per edge) ---------------------
__global__ void scatter_add(const float* __restrict__ xin, const int* __restrict__ src,
                            const int* __restrict__ dst, float* __restrict__ agg,
                            int E, int C) {
    const int wpb  = blockDim.x >> 5;
    const int lane = threadIdx.x & 31;
    int w  = blockIdx.x * wpb + (threadIdx.x >> 5);
    const int nW = gridDim.x * wpb;
    for (int e = w; e < E; e += nW) {
        const int s = src[e];
        const int d = dst[e];
        const float* xr = xin + (size_t)s * C;
        float*       ar = agg + (size_t)d * C;
        for (int c = lane; c < C; c += 32)
            atomicAdd(&ar[c], xr[c]);
    }
}

// -------------------- dual GEMM: out = (agg*invdeg)@Wl + bl + xin@Wr -------
// One block = one 16-row M tile; wave w = 16-col N tile (Co=128 -> 8 waves).
// A frag (16x4 f32): lanes 0-15 -> M rows, VGPR/halfwave -> K (per ISA 7.12.2).
// B frag (4x16 f32): lanes -> N cols, VGPR/halfwave -> K (mirror of A).
__global__ __launch_bounds__(256)
void sage_gemm(const float* __restrict__ agg, const float* __restrict__ xin,
               const float* __restrict__ invdeg,
               const float* __restrict__ Wl, const float* __restrict__ bl,
               const float* __restrict__ Wr,
               float* __restrict__ out, int K, int Co) {
    const int m0   = blockIdx.x << 4;
    const int wave = threadIdx.x >> 5;      // 0..7
    const int lane = threadIdx.x & 31;
    const int half = lane >> 4;             // 0 or 1
    const int l15  = lane & 15;
    const int n0   = wave << 4;
    const int row  = m0 + l15;
    const int col  = n0 + l15;
    const float scale = invdeg[row];

    const float* aRow = agg + (size_t)row * K + 2 * half;
    const float* xRow = xin + (size_t)row * K + 2 * half;

    v8f c = {};
    for (int k0 = 0; k0 < K; k0 += 4) {
        v2f a, b;
        // (agg * invdeg) @ Wl
        a.x = aRow[k0]     * scale;
        a.y = aRow[k0 + 1] * scale;
        b.x = Wl[(size_t)(k0 + 2 * half)     * Co + col];
        b.y = Wl[(size_t)(k0 + 2 * half + 1) * Co + col];
        c = __builtin_amdgcn_wmma_f32_16x16x4_f32(false, a, false, b, (short)0, c, false, false);
        // xin @ Wr
        a.x = xRow[k0];
        a.y = xRow[k0 + 1];
        b.x = Wr[(size_t)(k0 + 2 * half)     * Co + col];
        b.y = Wr[(size_t)(k0 + 2 * half + 1) * Co + col];
        c = __builtin_amdgcn_wmma_f32_16x16x4_f32(false, a, false, b, (short)0, c, false, false);
    }

    // C/D layout: lane holds (M = m0 + r + 8*half, N = col), r = VGPR index.
    const float bias = bl[col];
#pragma unroll
    for (int r = 0; r < 8; ++r) {
        const int m = m0 + r + 8 * half;
        out[(size_t)m * Co + col] = c[r] + bias;
    }
}

// -------------------- BatchNorm column stats (sum, sumsq) ------------------
__global__ void bn_stats(const float* __restrict__ h, float* __restrict__ stats,
                         int N, int C) {
    const int c   = threadIdx.x & (C - 1);
    const int rpb = blockDim.x / C;               // rows per block pass
    int r = blockIdx.x * rpb + threadIdx.x / C;
    const int stride = gridDim.x * rpb;
    float s = 0.f, s2 = 0.f;
    for (; r < N; r += stride) {
        const float v = h[(size_t)r * C + c];
        s += v; s2 += v * v;
    }
    atomicAdd(&stats[c], s);
    atomicAdd(&stats[C + c], s2);
}

__global__ void bn_apply_relu(float* __restrict__ h, const float* __restrict__ stats,
                              const float* __restrict__ g, const float* __restrict__ be,
                              int N, int C) {
    size_t i = (size_t)blockIdx.x * blockDim.x + threadIdx.x;
    const size_t total  = (size_t)N * C;
    const size_t stride = (size_t)gridDim.x * blockDim.x;
    const float invN = 1.0f / (float)N;
    for (; i < total; i += stride) {
        const int c = (int)(i & (C - 1));
        const float mu  = stats[c] * invN;
        const float var = stats[C + c] * invN - mu * mu;   // biased var (matches ref)
        float v = (h[i] - mu) * rsqrtf(var + 1e-5f) * g[c] + be[c];
        h[i] = fmaxf(v, 0.f);
    }
}

// -------------------- layer 2 (128->1) + Linear(1,1) + sigmoid -------------
__global__ void final_kernel(const float* __restrict__ agg, const float* __restrict__ hin,
                             const float* __restrict__ invdeg,
                             const float* __restrict__ Wl2, const float* __restrict__ bl2,
                             const float* __restrict__ Wr2,
                             const float* __restrict__ wF, const float* __restrict__ bF,
                             float* __restrict__ out, int N, int C) {
    const int wpb  = blockDim.x >> 5;
    const int lane = threadIdx.x & 31;
    int w = blockIdx.x * wpb + (threadIdx.x >> 5);
    const int nW = gridDim.x * wpb;
    for (int i = w; i < N; i += nW) {
        const float inv = invdeg[i];
        const float* ar = agg + (size_t)i * C;
        const float* hr = hin + (size_t)i * C;
        float s = 0.f;
        for (int c = lane; c < C; c += 32)
            s += ar[c] * inv * Wl2[c] + hr[c] * Wr2[c];
        // wave32 reduction
        for (int off = 16; off >= 1; off >>= 1) s += __shfl_xor(s, off, 32);
        if (lane == 0) {
            const float hval = s + bl2[0];
            const float z = hval * wF[0] + bF[0];
            out[i] = 1.0f / (1.0f + expf(-z));
        }
    }
}

// ---------------------------------------------------------------------------
extern "C" void kernel_launch(void* const* d_in, const int* in_sizes, int n_in,
                              void* d_out, int out_size, void* d_ws, size_t ws_size,
                              hipStream_t stream) {
    (void)n_in; (void)out_size; (void)ws_size;

    const float* x   = (const float*)d_in[0];
    const int*   ei  = (const int*)d_in[1];
    const float* Wl0 = (const float*)d_in[2];
    const float* bl0 = (const float*)d_in[3];
    const float* Wr0 = (const float*)d_in[4];
    const float* Wl1 = (const float*)d_in[5];
    const float* bl1 = (const float*)d_in[6];
    const float* Wr1 = (const float*)d_in[7];
    const float* Wl2 = (const float*)d_in[8];
    const float* bl2 = (const float*)d_in[9];
    const float* Wr2 = (const float*)d_in[10];
    const float* g0  = (const float*)d_in[11];
    const float* be0 = (const float*)d_in[12];
    const float* g1  = (const float*)d_in[13];
    const float* be1 = (const float*)d_in[14];
    const float* wF  = (const float*)d_in[15];
    const float* bF  = (const float*)d_in[16];

    const int N = N_NODES;
    const int E = in_sizes[1] / 2;          // edge_index is [2, E]
    const int* src = ei;
    const int* dst = ei + E;

    // workspace layout (floats)
    float* ws     = (float*)d_ws;
    float* invdeg = ws;                              // N
    float* agg    = invdeg + N;                      // N*128 (layer scratch)
    float* hA     = agg + (size_t)N * 128;           // N*128
    float* hB     = hA  + (size_t)N * 128;           // N*128
    float* stats  = hB  + (size_t)N * 128;           // 256

    // degrees (shared by all layers)
    hipMemsetAsync(invdeg, 0, (size_t)N * sizeof(float), stream);
    degree_kernel<<<2048, 256, 0, stream>>>(dst, invdeg, E);
    invdeg_kernel<<<(N + 255) / 256, 256, 0, stream>>>(invdeg, N);

    // ---- layer 0: 64 -> 128 ----
    hipMemsetAsync(agg, 0, (size_t)N * 64 * sizeof(float), stream);
    scatter_add<<<4096, 256, 0, stream>>>(x, src, dst, agg, E, 64);
    sage_gemm<<<N / 16, 256, 0, stream>>>(agg, x, invdeg, Wl0, bl0, Wr0, hA, 64, 128);
    hipMemsetAsync(stats, 0, 256 * sizeof(float), stream);
    bn_stats<<<512, 256, 0, stream>>>(hA, stats, N, 128);
    bn_apply_relu<<<2048, 256, 0, stream>>>(hA, stats, g0, be0, N, 128);

    // ---- layer 1: 128 -> 128 ----
    hipMemsetAsync(agg, 0, (size_t)N * 128 * sizeof(float), stream);
    scatter_add<<<4096, 256, 0, stream>>>(hA, src, dst, agg, E, 128);
    sage_gemm<<<N / 16, 256, 0, stream>>>(agg, hA, invdeg, Wl1, bl1, Wr1, hB, 128, 128);
    hipMemsetAsync(stats, 0, 256 * sizeof(float), stream);
    bn_stats<<<512, 256, 0, stream>>>(hB, stats, N, 128);
    bn_apply_relu<<<2048, 256, 0, stream>>>(hB, stats, g1, be1, N, 128);

    // ---- layer 2: 128 -> 1, then Linear(1,1) + sigmoid ----
    hipMemsetAsync(agg, 0, (size_t)N * 128 * sizeof(float), stream);
    scatter_add<<<4096, 256, 0, stream>>>(hB, src, dst, agg, E, 128);
    final_kernel<<<2048, 256, 0, stream>>>(agg, hB, invdeg, Wl2, bl2, Wr2, wF, bF,
                                           (float*)d_out, N, 128);
}